// position_attention_88897233093026
// MI455X (gfx1250) — compile-verified
//
#include <hip/hip_runtime.h>
#include <hip/hip_bf16.h>
#include <stdint.h>

// ---------------------------------------------------------------------------
// MI455X / gfx1250 implementation of the bidirectional position-attention net.
//   B=16, S=128, E=128, OUT=5.
// GEMMs use V_WMMA_F32_16X16X4_F32 (fp32 matrix pipe, exact precision).
// The [B,i,j,E] attention tensor is never materialized: tanh/exp/softmax/
// weighted-sum are fused into one j-loop (tanh output bounded => no max pass).
// ---------------------------------------------------------------------------

typedef __attribute__((ext_vector_type(2))) float v2f;
typedef __attribute__((ext_vector_type(8))) float v8f;

#define SEQ   128
#define EMB   128
#define BATCH 16
#define NROWS (BATCH * SEQ)       // 2048 rows in all [*,E] activations
#define SLAB  (NROWS * EMB)       // 262144 floats = 1 MB per activation slab

#define EP_NONE 0
#define EP_SELU 1
#define EP_BIAS 2
#define EP_RELU 3

__device__ __forceinline__ float fast_tanh(float x) {
#if __has_builtin(__builtin_amdgcn_tanhf)
    return __builtin_amdgcn_tanhf(x);       // v_tanh_f32 (gfx1250 TRANS op)
#else
    float e = __expf(2.0f * x);             // v_exp_f32 path
    return 1.0f - 2.0f / (e + 1.0f);
#endif
}

// ---------------------------------------------------------------------------
// we[row,e] = Wemb[word[row], e] + Pemb[pos[row], e]
// ---------------------------------------------------------------------------
__global__ __launch_bounds__(EMB) void k_embed(const int* __restrict__ word,
                                               const int* __restrict__ pos,
                                               const float* __restrict__ Wemb,
                                               const float* __restrict__ Pemb,
                                               float* __restrict__ we) {
    int row = blockIdx.x;
    int e   = threadIdx.x;
    int w   = word[row];
    int p   = pos[row];
    we[row * EMB + e] = Wemb[w * EMB + e] + Pemb[p * EMB + e];
}

// ---------------------------------------------------------------------------
// C[2048,128] = epilogue( A[2048,128] @ W[128,128] + bias )
// Block = 128 threads = 4 waves; wave w computes rows [blk*64+w*16, +16) x all
// 128 cols as 8 WMMA 16x16 tiles, K-loop of 32 x V_WMMA_F32_16X16X4_F32.
// W staged transposed in LDS (pad 130) so B-fragments are single ds_load_b64.
// A 16x4 f32 fragment layout (ISA 7.12.2): lane<16 -> K={0,1}, lane>=16 ->
// K={2,3}, row = lane&15.  B mirrors with N = lane&15.
// D layout: VGPR r -> M = (lane>>4)*8 + r, N = lane&15 within tile.
// ---------------------------------------------------------------------------
__global__ __launch_bounds__(128) void k_gemm_e128(const float* __restrict__ A,
                                                   const float* __restrict__ W,
                                                   const float* __restrict__ bias,
                                                   float* __restrict__ C,
                                                   int mode) {
    __shared__ float Wt[EMB * 130];
    int tid = threadIdx.x;
    for (int idx = tid; idx < EMB * EMB; idx += 128) {
        int k = idx >> 7;
        int c = idx & 127;
        Wt[c * 130 + k] = W[idx];            // coalesced read, transposed store
    }
    __syncthreads();

    int wave    = tid >> 5;
    int lane    = tid & 31;
    int rowbase = blockIdx.x * 64 + wave * 16;
    int m       = lane & 15;
    int kk      = (lane >> 4) * 2;

    v8f acc[8];
#pragma unroll
    for (int t = 0; t < 8; ++t) acc[t] = (v8f){0.f,0.f,0.f,0.f,0.f,0.f,0.f,0.f};

    const float* Arow = A + (rowbase + m) * EMB;
#pragma unroll 4
    for (int ks = 0; ks < 32; ++ks) {
        int k0 = ks * 4 + kk;
        v2f a;
        a.x = Arow[k0];
        a.y = Arow[k0 + 1];
#pragma unroll
        for (int t = 0; t < 8; ++t) {
            int n = t * 16 + m;
            v2f b = *(const v2f*)(&Wt[n * 130 + k0]);   // ds_load_b64
            acc[t] = __builtin_amdgcn_wmma_f32_16x16x4_f32(
                false, a, false, b, (short)0, acc[t], false, false);
        }
    }

    int rbase = rowbase + (lane >> 4) * 8;
#pragma unroll
    for (int t = 0; t < 8; ++t) {
        int col  = t * 16 + m;
        float bv = (mode == EP_NONE) ? 0.0f : bias[col];
#pragma unroll
        for (int r = 0; r < 8; ++r) {
            float v = acc[t][r] + bv;
            if (mode == EP_SELU) {
                v = (v >= 0.0f)
                        ? 1.0507009873554805f * v
                        : 1.0507009873554805f * 1.6732632423543772f *
                              (__expf(v) - 1.0f);
            } else if (mode == EP_RELU) {
                v = v > 0.0f ? v : 0.0f;
            }
            C[(rbase + r) * EMB + col] = v;
        }
    }
}

// ---------------------------------------------------------------------------
// Fused attention: for each (b,i,e), over masked j:
//   w_j = exp(5*tanh((d2[b,j,e]+d3[b,i,e])/5))   (bias pre-folded into d2)
//   out = sum_j w_j * we1[b,j,e] / sum_j w_j     (0 if no masked j)
// tanh output in [-5,5] => exp never overflows => single pass, no max.
// Block = 128 threads (one per e); mask test is wave-uniform (scalar branch).
// dir=0: j<i (fwd, tril -1).  dir=1: j>i (bwd, triu +1).
// ---------------------------------------------------------------------------
__global__ __launch_bounds__(EMB) void k_attn(const float* __restrict__ d2,
                                              const float* __restrict__ d3,
                                              const float* __restrict__ we1,
                                              const unsigned char* __restrict__ smask,
                                              float* __restrict__ outp,
                                              int dir) {
    int b = blockIdx.x >> 7;
    int i = blockIdx.x & 127;
    int e = threadIdx.x;

    const float* d2b = d2 + b * SEQ * EMB;
    const float* wb  = we1 + b * SEQ * EMB;
    float d3v        = d3[(b * SEQ + i) * EMB + e];
    const unsigned char* mrow = smask + (b * SEQ + i) * SEQ;

    int jlo = dir ? (i + 1) : 0;
    int jhi = dir ? SEQ : i;

    float den = 0.0f, num = 0.0f;
    for (int j = jlo; j < jhi; ++j) {
        if (mrow[j]) {                               // uniform across the wave
            float x = (d2b[j * EMB + e] + d3v) * 0.2f;
            float w = __expf(5.0f * fast_tanh(x));
            den += w;
            num += w * wb[j * EMB + e];
        }
    }
    outp[(b * SEQ + i) * EMB + e] = (den > 0.0f) ? (num / den) : 0.0f;
}

// ---------------------------------------------------------------------------
// g = sigmoid(t1 + t2);  fused = g*f_res + (1-g)*b_res
// ---------------------------------------------------------------------------
__global__ __launch_bounds__(256) void k_fuse(const float* __restrict__ t1,
                                              const float* __restrict__ t2,
                                              const float* __restrict__ fr,
                                              const float* __restrict__ br,
                                              float* __restrict__ fused) {
    int idx = blockIdx.x * 256 + threadIdx.x;
    float g = 1.0f / (1.0f + __expf(-(t1[idx] + t2[idx])));
    fused[idx] = g * fr[idx] + (1.0f - g) * br[idx];
}

// ---------------------------------------------------------------------------
// out[b, 0..4] = h[b, :].flatten() @ W5[16384, 5]
// One block per batch row; LDS tree reduction. N=5 is too thin for WMMA.
// ---------------------------------------------------------------------------
__global__ __launch_bounds__(256) void k_final(const float* __restrict__ h,
                                               const float* __restrict__ W5,
                                               float* __restrict__ out) {
    __shared__ float red[5][256];
    int b   = blockIdx.x;
    int tid = threadIdx.x;
    float acc[5] = {0.f, 0.f, 0.f, 0.f, 0.f};
    const float* hb = h + b * (SEQ * EMB);
    for (int k = tid; k < SEQ * EMB; k += 256) {
        float hv = hb[k];
        const float* wr = W5 + k * 5;
#pragma unroll
        for (int o = 0; o < 5; ++o) acc[o] += hv * wr[o];
    }
#pragma unroll
    for (int o = 0; o < 5; ++o) red[o][tid] = acc[o];
    __syncthreads();
    for (int s = 128; s > 0; s >>= 1) {
        if (tid < s) {
#pragma unroll
            for (int o = 0; o < 5; ++o) red[o][tid] += red[o][tid + s];
        }
        __syncthreads();
    }
    if (tid < 5) out[b * 5 + tid] = red[tid][0];
}

// ---------------------------------------------------------------------------
extern "C" void kernel_launch(void* const* d_in, const int* in_sizes, int n_in,
                              void* d_out, int out_size, void* d_ws, size_t ws_size,
                              hipStream_t stream) {
    const int*   word  = (const int*)d_in[0];
    const int*   pos   = (const int*)d_in[1];
    const unsigned char* smask = (const unsigned char*)d_in[2];   // bool [B,S,S]
    const float* Wemb  = (const float*)d_in[3];
    const float* Pemb  = (const float*)d_in[4];
    const float* W1f   = (const float*)d_in[5];
    const float* b1f   = (const float*)d_in[6];
    const float* W2f   = (const float*)d_in[7];
    const float* W3f   = (const float*)d_in[8];
    const float* biasf = (const float*)d_in[9];
    const float* W1b   = (const float*)d_in[10];
    const float* b1b   = (const float*)d_in[11];
    const float* W2b   = (const float*)d_in[12];
    const float* W3b   = (const float*)d_in[13];
    const float* biasb = (const float*)d_in[14];
    const float* Wg1   = (const float*)d_in[15];
    const float* bg1   = (const float*)d_in[16];
    const float* Wg2   = (const float*)d_in[17];
    const float* bg2   = (const float*)d_in[18];
    const float* W4    = (const float*)d_in[19];
    const float* b4    = (const float*)d_in[20];
    const float* W5    = (const float*)d_in[21];
    float* out = (float*)d_out;

    float* ws = (float*)d_ws;
    float* we    = ws + 0 * SLAB;
    float* we1f  = ws + 1 * SLAB;
    float* we1b  = ws + 2 * SLAB;
    float* d2f   = ws + 3 * SLAB;   // reused later as t1
    float* d3f   = ws + 4 * SLAB;   // reused later as t2
    float* d2b   = ws + 5 * SLAB;   // reused later as fused
    float* d3b   = ws + 6 * SLAB;   // reused later as h
    float* fres  = ws + 7 * SLAB;
    float* bres  = ws + 8 * SLAB;

    const dim3 gEmb(NROWS), bEmb(EMB);
    const dim3 gGemm(NROWS / 64), bGemm(128);
    const dim3 gAttn(BATCH * SEQ), bAttn(EMB);

    // Embeddings
    k_embed<<<gEmb, bEmb, 0, stream>>>(word, pos, Wemb, Pemb, we);

    // we1 = selu(we @ W1 + b1), both directions
    k_gemm_e128<<<gGemm, bGemm, 0, stream>>>(we, W1f, b1f, we1f, EP_SELU);
    k_gemm_e128<<<gGemm, bGemm, 0, stream>>>(we, W1b, b1b, we1b, EP_SELU);

    // d2 = we1 @ W2 + bias (logit bias folded here); d3 = we1 @ W3
    k_gemm_e128<<<gGemm, bGemm, 0, stream>>>(we1f, W2f, biasf, d2f, EP_BIAS);
    k_gemm_e128<<<gGemm, bGemm, 0, stream>>>(we1f, W3f, nullptr, d3f, EP_NONE);
    k_gemm_e128<<<gGemm, bGemm, 0, stream>>>(we1b, W2b, biasb, d2b, EP_BIAS);
    k_gemm_e128<<<gGemm, bGemm, 0, stream>>>(we1b, W3b, nullptr, d3b, EP_NONE);

    // Fused masked-softmax attention, both directions
    k_attn<<<gAttn, bAttn, 0, stream>>>(d2f, d3f, we1f, smask, fres, 0);
    k_attn<<<gAttn, bAttn, 0, stream>>>(d2b, d3b, we1b, smask, bres, 1);

    // Gate: t1 = fres@Wg1+bg1, t2 = bres@Wg2+bg2; fused = g*f + (1-g)*b
    float* t1 = d2f;
    float* t2 = d3f;
    float* fused = d2b;
    k_gemm_e128<<<gGemm, bGemm, 0, stream>>>(fres, Wg1, bg1, t1, EP_BIAS);
    k_gemm_e128<<<gGemm, bGemm, 0, stream>>>(bres, Wg2, bg2, t2, EP_BIAS);
    k_fuse<<<dim3(SLAB / 256), dim3(256), 0, stream>>>(t1, t2, fres, bres, fused);

    // h = relu(fused @ W4 + b4);  out = h.reshape(B,-1) @ W5
    float* h = d3b;
    k_gemm_e128<<<gGemm, bGemm, 0, stream>>>(fused, W4, b4, h, EP_RELU);
    k_final<<<dim3(BATCH), dim3(256), 0, stream>>>(h, W5, out);
}